// CacheOnlyAttentionLayer_65360812310567
// MI455X (gfx1250) — compile-verified
//
#include <hip/hip_runtime.h>
#include <cstdint>
#include <cstddef>

// ---------------------------------------------------------------------------
// basic_cache / reshape_and_cache scatter for MI455X (gfx1250)
//
//   to_cache     : [65536, 8, 128] f32   (4096 B per token row)
//   kv_cache     : [8192, 16, 8, 128] f32 (131072 slots x 4096 B)
//   slot_mapping : [65536] i32 (unique)
//   out          : updated kv_cache (full 512 MB)
//
// Strategy: inverse-map merge.
//   Pass A: zero inv[131072] in d_ws
//   Pass B: inv[slot_mapping[t]] = t + 1            (RT: re-read by pass C)
//   Pass C: for every slot, stream 4 KB from (token row | old cache row)
//           into out — fully sequential writes — using CDNA5 async
//           global<->LDS DMA (ASYNCcnt), double-buffered per wave, with
//           non-temporal hints so the 1 GB single-pass stream does not
//           thrash the 192 MB L2.
// Traffic ~1.03 GB vs ~1.75 GB naive -> ~45 us floor at 23.3 TB/s.
// ---------------------------------------------------------------------------

#define NUM_TOKENS   65536
#define NUM_HEADS    8
#define HEAD_SIZE    128
#define ROW_FLOATS   (NUM_HEADS * HEAD_SIZE)      // 1024 floats
#define ROW_BYTES    (ROW_FLOATS * 4)             // 4096 bytes
#define NUM_PBLOCKS  8192
#define PBLOCK_SIZE  16
#define NSLOTS       (NUM_PBLOCKS * PBLOCK_SIZE)  // 131072 slots

typedef float v4f __attribute__((ext_vector_type(4)));
typedef int   v4i __attribute__((ext_vector_type(4)));

// ---------------------------------------------------------------------------
// CDNA5 async global<->LDS helpers. One wave32 moves 512 B per instruction
// (16 B per lane). The signed 24-bit IOFFSET is added to BOTH the per-lane
// LDS address and the per-lane global address (CDNA5 ISA),
// so one (lds, global) address pair covers a whole 4 KB slot. TH=NT: the
// streams are touched exactly once.
// ---------------------------------------------------------------------------
__device__ __forceinline__ void slot_load_async(unsigned lds, const void* g) {
  asm volatile("global_load_async_to_lds_b128 %0, %1, off th:TH_LOAD_NT"             :: "v"(lds), "v"(g) : "memory");
  asm volatile("global_load_async_to_lds_b128 %0, %1, off offset:512 th:TH_LOAD_NT"  :: "v"(lds), "v"(g) : "memory");
  asm volatile("global_load_async_to_lds_b128 %0, %1, off offset:1024 th:TH_LOAD_NT" :: "v"(lds), "v"(g) : "memory");
  asm volatile("global_load_async_to_lds_b128 %0, %1, off offset:1536 th:TH_LOAD_NT" :: "v"(lds), "v"(g) : "memory");
  asm volatile("global_load_async_to_lds_b128 %0, %1, off offset:2048 th:TH_LOAD_NT" :: "v"(lds), "v"(g) : "memory");
  asm volatile("global_load_async_to_lds_b128 %0, %1, off offset:2560 th:TH_LOAD_NT" :: "v"(lds), "v"(g) : "memory");
  asm volatile("global_load_async_to_lds_b128 %0, %1, off offset:3072 th:TH_LOAD_NT" :: "v"(lds), "v"(g) : "memory");
  asm volatile("global_load_async_to_lds_b128 %0, %1, off offset:3584 th:TH_LOAD_NT" :: "v"(lds), "v"(g) : "memory");
}

__device__ __forceinline__ void slot_store_async(const void* g, unsigned lds) {
  asm volatile("global_store_async_from_lds_b128 %0, %1, off th:TH_STORE_NT"             :: "v"(g), "v"(lds) : "memory");
  asm volatile("global_store_async_from_lds_b128 %0, %1, off offset:512 th:TH_STORE_NT"  :: "v"(g), "v"(lds) : "memory");
  asm volatile("global_store_async_from_lds_b128 %0, %1, off offset:1024 th:TH_STORE_NT" :: "v"(g), "v"(lds) : "memory");
  asm volatile("global_store_async_from_lds_b128 %0, %1, off offset:1536 th:TH_STORE_NT" :: "v"(g), "v"(lds) : "memory");
  asm volatile("global_store_async_from_lds_b128 %0, %1, off offset:2048 th:TH_STORE_NT" :: "v"(g), "v"(lds) : "memory");
  asm volatile("global_store_async_from_lds_b128 %0, %1, off offset:2560 th:TH_STORE_NT" :: "v"(g), "v"(lds) : "memory");
  asm volatile("global_store_async_from_lds_b128 %0, %1, off offset:3072 th:TH_STORE_NT" :: "v"(g), "v"(lds) : "memory");
  asm volatile("global_store_async_from_lds_b128 %0, %1, off offset:3584 th:TH_STORE_NT" :: "v"(g), "v"(lds) : "memory");
}

__device__ __forceinline__ void wait_async0() {
  asm volatile("s_wait_asynccnt 0x0" ::: "memory");
}

// ---------------------------------------------------------------------------
// Pass A: zero the inverse map (131072 i32 = 32768 int4). 128 blocks x 256.
// ---------------------------------------------------------------------------
__global__ __launch_bounds__(256) void kv_zero_inv(v4i* __restrict__ inv4) {
  const int i = blockIdx.x * 256 + threadIdx.x;
  v4i z = {0, 0, 0, 0};
  __builtin_nontemporal_store(z, &inv4[i]);
}

// ---------------------------------------------------------------------------
// Pass B: inv[slot_mapping[t]] = t + 1. 256 blocks x 256.
// Regular-temporal stores on purpose: pass C re-reads inv from L2.
// ---------------------------------------------------------------------------
__global__ __launch_bounds__(256) void kv_scatter_inv(const int* __restrict__ map,
                                                      int* __restrict__ inv) {
  const int t = blockIdx.x * 256 + threadIdx.x;
  inv[map[t]] = t + 1;
}

// ---------------------------------------------------------------------------
// Pass C: merge. One slot (4 KB) per wave per iteration; 2048 blocks x 8
// waves = 16384 waves -> exactly 8 slots per wave. Each wave owns two 4 KB
// LDS buffers (8 waves * 8 KB = 64 KB static LDS) and double-buffers:
//   wait(asynccnt==0)  -> loads into `cur` landed (and prior stores drained)
//   issue stores(cur)  -> overlap with
//   issue loads(nxt)
// All control flow is wave-uniform, so EXEC stays all-ones for the async ops.
// ---------------------------------------------------------------------------
__global__ __launch_bounds__(256) void kv_merge(const float* __restrict__ toc,
                                                const float* __restrict__ kvc,
                                                const int* __restrict__ inv,
                                                float* __restrict__ out) {
  __shared__ __align__(16) unsigned char smem[8 * 2 * 4096];  // 64 KB

  const int lane  = threadIdx.x & 31;
  const int wv    = threadIdx.x >> 5;
  const int gwave = blockIdx.x * 8 + wv;
  const int nwave = gridDim.x * 8;

  // Flat LDS pointer: low 32 bits == LDS byte offset (aperture rule).
  const unsigned lbase =
      (unsigned)(uintptr_t)smem + (unsigned)(wv * 8192) + (unsigned)(lane * 16);
  unsigned cur = lbase;           // buffer 0
  unsigned nxt = lbase + 4096u;   // buffer 1

  int slot = gwave;
  if (slot < NSLOTS) {
    const int t = inv[slot];
    const float* src = t ? (toc + (size_t)(t - 1) * ROW_FLOATS)
                         : (kvc + (size_t)slot * ROW_FLOATS);
    slot_load_async(cur, (const char*)src + lane * 16);
  }

  while (slot < NSLOTS) {
    const int nslot = slot + nwave;
    const float* nsrc = nullptr;
    if (nslot < NSLOTS) {                         // wave-uniform branch
      const int t = inv[nslot];                   // overlaps with in-flight DMA
      nsrc = t ? (toc + (size_t)(t - 1) * ROW_FLOATS)
               : (kvc + (size_t)nslot * ROW_FLOATS);
    }

    wait_async0();  // loads into `cur` complete; previous stores drained

    slot_store_async((char*)out + (size_t)slot * ROW_BYTES + lane * 16, cur);
    if (nslot < NSLOTS) {
      slot_load_async(nxt, (const char*)nsrc + lane * 16);
    }

    const unsigned tmp = cur; cur = nxt; nxt = tmp;
    slot = nslot;
  }
  wait_async0();  // drain final stores before LDS goes out of scope
}

// ---------------------------------------------------------------------------
// Fallback (only if ws_size < 512 KB): plain copy + scatter through VGPRs
// with 128-bit nontemporal accesses. ~1.75 GB of traffic.
// ---------------------------------------------------------------------------
__global__ __launch_bounds__(256) void kv_copy_all(const v4f* __restrict__ in,
                                                   v4f* __restrict__ out, long n4) {
  long i = (long)blockIdx.x * 256 + threadIdx.x;
  const long stride = (long)gridDim.x * 256;
  for (; i < n4; i += stride) {
    v4f v = __builtin_nontemporal_load(&in[i]);
    __builtin_nontemporal_store(v, &out[i]);
  }
}

__global__ __launch_bounds__(256) void kv_scatter_rows(const v4f* __restrict__ toc,
                                                       const int* __restrict__ map,
                                                       v4f* __restrict__ out) {
  const long total = (long)NUM_TOKENS * (ROW_FLOATS / 4);  // 16,777,216 float4
  long i = (long)blockIdx.x * 256 + threadIdx.x;
  const long stride = (long)gridDim.x * 256;
  for (; i < total; i += stride) {
    const int token = (int)(i >> 8);   // 256 float4 per row
    const int e     = (int)(i & 255);
    const int slot  = map[token];
    v4f v = __builtin_nontemporal_load(&toc[i]);
    __builtin_nontemporal_store(v, &out[(size_t)slot * 256 + e]);
  }
}

// ---------------------------------------------------------------------------
// Launch
// ---------------------------------------------------------------------------
extern "C" void kernel_launch(void* const* d_in, const int* in_sizes, int n_in,
                              void* d_out, int out_size, void* d_ws, size_t ws_size,
                              hipStream_t stream) {
  const float* to_cache = (const float*)d_in[0];
  const float* kv_cache = (const float*)d_in[1];
  const int*   slot_map = (const int*)d_in[2];
  float*       out      = (float*)d_out;

  if (ws_size >= (size_t)NSLOTS * sizeof(int)) {
    int* inv = (int*)d_ws;
    kv_zero_inv   <<<NSLOTS / 4 / 256, 256, 0, stream>>>((v4i*)inv);
    kv_scatter_inv<<<NUM_TOKENS / 256, 256, 0, stream>>>(slot_map, inv);
    kv_merge      <<<2048,             256, 0, stream>>>(to_cache, kv_cache, inv, out);
  } else {
    const long n4 = (long)out_size / 4;
    kv_copy_all    <<<4096, 256, 0, stream>>>((const v4f*)kv_cache, (v4f*)out, n4);
    kv_scatter_rows<<<4096, 256, 0, stream>>>((const v4f*)to_cache, slot_map, (v4f*)out);
  }
}